// AdderNet_30760555774423
// MI455X (gfx1250) — compile-verified
//
#include <hip/hip_runtime.h>
#include <hip/hip_bf16.h>

typedef __attribute__((ext_vector_type(2))) float v2f;
typedef __attribute__((ext_vector_type(8))) float v8f;

#define BN_EPS 1e-5f

// ---------------------------------------------------------------------------
// Adder "conv": out[b,o,y,x] = -sum_{c,ky,kx} | in_patch - w |  (zero padding
// contributes |0 - w|). OT=4 output channels per block: each global patch
// load feeds 4 |diff| accumulations (abs folds into v_add src modifier).
// Weights for the 4 channels staged in LDS (wave-uniform broadcast reads).
// ---------------------------------------------------------------------------
template <int OT>
__global__ void adder_conv(const float* __restrict__ in, const float* __restrict__ w,
                           float* __restrict__ out, int Cin, int H, int W,
                           int Cout, int kh, int kw, int stride, int pad,
                           int Ho, int Wo, int woShift) {
    __shared__ float sw[OT * 576]; // max Cin*kh*kw = 64*9
    const int nOG = Cout / OT;
    const int og = blockIdx.x % nOG;
    const int b  = blockIdx.x / nOG;
    const int o0 = og * OT;
    const int K  = Cin * kh * kw;
    for (int i = threadIdx.x; i < K * OT; i += blockDim.x) sw[i] = w[o0 * K + i];
    __syncthreads();
    const int HWo  = Ho * Wo;
    const int wMask = Wo - 1;
    for (int pos = threadIdx.x; pos < HWo; pos += blockDim.x) {
        const int y = pos >> woShift, x = pos & wMask;
        const int iy0 = y * stride - pad, ix0 = x * stride - pad;
        float acc[OT];
#pragma unroll
        for (int j = 0; j < OT; ++j) acc[j] = 0.0f;
        for (int c = 0; c < Cin; ++c) {
            const float* ib = in + (long)(b * Cin + c) * H * W;
            const float* wb = sw + c * kh * kw;
            for (int ky = 0; ky < kh; ++ky) {
                const int iy = iy0 + ky;
                const bool yok = ((unsigned)iy < (unsigned)H);
                for (int kx = 0; kx < kw; ++kx) {
                    const int ix = ix0 + kx;
                    float v = (yok && (unsigned)ix < (unsigned)W) ? ib[iy * W + ix] : 0.0f;
                    const int k = ky * kw + kx;
#pragma unroll
                    for (int j = 0; j < OT; ++j)
                        acc[j] += fabsf(v - wb[j * K + k]);
                }
            }
        }
#pragma unroll
        for (int j = 0; j < OT; ++j)
            out[(long)(b * Cout + o0 + j) * HWo + pos] = -acc[j];
    }
}

// ---------------------------------------------------------------------------
// BN batch statistics on the matrix pipe: per-channel Sum / SumSq via
// V_WMMA_F32_16X16X4_F32 with B = ones (exact f32 row-sum reduction).
// A layout (16x4 f32): lanes 0-15 -> row M=lane, K={0,1}; lanes 16-31 ->
// row M=lane-16, K={2,3}. Grid = (C/16 groups) x (8 batch slices); each
// wave streams one batch row with pure pointer-increment addressing.
// Partials land in `part`; a tiny kernel reduces them in fixed order
// (deterministic, no float atomics).
// ---------------------------------------------------------------------------
__global__ void bn_stats_wmma_vec(const float* __restrict__ in, float* __restrict__ part,
                                  int C, int hwShift) {
    __shared__ float lds[8][2][16];
    const int tid  = threadIdx.x;
    const int wave = tid >> 5;
    const int lane = tid & 31;
    const int cl   = lane & 15;
    const int khi  = lane >> 4;                  // 0: K={0,1}, 1: K={2,3}
    const int g    = blockIdx.x;                 // channel group (C multiple of 16)
    const int s    = blockIdx.y;                 // batch slice (8 slices)
    const int c    = g * 16 + cl;
    const int b    = s * 8 + wave;               // one batch row per wave (B=64)
    const float* p = in + (((long)(b * C + c)) << hwShift) + khi * 2;
    const int nCh  = (1 << hwShift) >> 2;        // HW/4 chunks per row
    v8f accS = {}; v8f accQ = {};
    const v2f ones = {1.0f, 1.0f};
#pragma unroll 4
    for (int t = 0; t < nCh; ++t) {
        v2f a = *(const v2f*)(p + t * 4);        // global_load_b64
        v2f q = {a.x * a.x, a.y * a.y};
        accS = __builtin_amdgcn_wmma_f32_16x16x4_f32(false, a, false, ones,
                                                     (short)0, accS, false, false);
        accQ = __builtin_amdgcn_wmma_f32_16x16x4_f32(false, q, false, ones,
                                                     (short)0, accQ, false, false);
    }
    if (lane == 0 || lane == 16) {
#pragma unroll
        for (int j = 0; j < 8; ++j) {
            lds[wave][0][khi * 8 + j] = accS[j];
            lds[wave][1][khi * 8 + j] = accQ[j];
        }
    }
    __syncthreads();
    if (tid < 16) {
        float S = 0.0f, Q = 0.0f;
        for (int wv = 0; wv < 8; ++wv) { S += lds[wv][0][tid]; Q += lds[wv][1][tid]; }
        part[((g * 8 + s) * 2 + 0) * 16 + tid] = S;
        part[((g * 8 + s) * 2 + 1) * 16 + tid] = Q;
    }
}

// Small path (HW == 1, final BN over [B, C]): single block, WMMA reduction.
__global__ void bn_stats_wmma_small(const float* __restrict__ in, float* __restrict__ part,
                                    int C, int N) {
    __shared__ float lds[8][2][16];
    const int tid  = threadIdx.x;
    const int wave = tid >> 5;
    const int lane = tid & 31;
    const int cl   = lane & 15;
    const int khi  = lane >> 4;
    const int g    = blockIdx.x;
    const int c    = g * 16 + cl;
    const bool ok  = (c < C);
    const int cc   = ok ? c : (C - 1);
    v8f accS = {}; v8f accQ = {};
    const v2f ones = {1.0f, 1.0f};
    const int nChunks = N >> 2;                  // N divisible by 4
    for (int t = wave; t < nChunks; t += 8) {
        const int n0 = t * 4 + khi * 2;
        float v0 = in[n0 * C + cc];
        float v1 = in[(n0 + 1) * C + cc];
        v0 = ok ? v0 : 0.0f;
        v1 = ok ? v1 : 0.0f;
        v2f a = {v0, v1};
        v2f q = {v0 * v0, v1 * v1};
        accS = __builtin_amdgcn_wmma_f32_16x16x4_f32(false, a, false, ones,
                                                     (short)0, accS, false, false);
        accQ = __builtin_amdgcn_wmma_f32_16x16x4_f32(false, q, false, ones,
                                                     (short)0, accQ, false, false);
    }
    if (lane == 0 || lane == 16) {
#pragma unroll
        for (int j = 0; j < 8; ++j) {
            lds[wave][0][khi * 8 + j] = accS[j];
            lds[wave][1][khi * 8 + j] = accQ[j];
        }
    }
    __syncthreads();
    if (tid < 16) {
        float S = 0.0f, Q = 0.0f;
        for (int wv = 0; wv < 8; ++wv) { S += lds[wv][0][tid]; Q += lds[wv][1][tid]; }
        part[(g * 2 + 0) * 16 + tid] = S;
        part[(g * 2 + 1) * 16 + tid] = Q;
    }
}

// Deterministic fixed-order reduction of slice partials -> stats[0..C-1]=Sum,
// stats[64..64+C-1]=SumSq.
__global__ void bn_stats_reduce(const float* __restrict__ part, float* __restrict__ stats,
                                int C, int S) {
    const int c = threadIdx.x;
    if (c >= C) return;
    const int g = c >> 4, cl = c & 15;
    float Ssum = 0.0f, Q = 0.0f;
    for (int s = 0; s < S; ++s) {
        Ssum += part[((g * S + s) * 2 + 0) * 16 + cl];
        Q    += part[((g * S + s) * 2 + 1) * 16 + cl];
    }
    stats[c] = Ssum;
    stats[64 + c] = Q;
}

// ---------------------------------------------------------------------------
// Fused BN-apply (training-mode BN, biased variance) + optional residual/ReLU.
// mode 0: BN+ReLU; mode 1: BN only; mode 2: BN + residual + ReLU.
// ---------------------------------------------------------------------------
__global__ void bn_apply(const float* __restrict__ in, float* __restrict__ out,
                         const float* __restrict__ stats, const float* __restrict__ g,
                         const float* __restrict__ bb, int C, int cMask, int hwShift,
                         int mode, const float* __restrict__ res, float invN, int total) {
    const int idx = blockIdx.x * blockDim.x + threadIdx.x;
    if (idx >= total) return;
    const int c = (cMask >= 0) ? ((idx >> hwShift) & cMask) : ((idx >> hwShift) % C);
    const float mean = stats[c] * invN;
    const float var  = stats[64 + c] * invN - mean * mean;
    const float sc = g[c] * rsqrtf(var + BN_EPS);
    const float sh = bb[c] - mean * sc;
    float y = in[idx] * sc + sh;
    if (mode == 2) y += res[idx];
    if (mode != 1) y = fmaxf(y, 0.0f);
    out[idx] = y;
}

__global__ void avgpool_k(const float* __restrict__ in, float* __restrict__ out, int n) {
    const int i = blockIdx.x * blockDim.x + threadIdx.x;
    if (i >= n) return;
    const float4* p = (const float4*)(in + i * 64);
    float s = 0.0f;
#pragma unroll
    for (int j = 0; j < 16; ++j) { float4 v = p[j]; s += v.x + v.y + v.z + v.w; }
    out[i] = s * (1.0f / 64.0f);
}

__global__ void fc_adder(const float* __restrict__ pool, const float* __restrict__ w,
                         float* __restrict__ out) {
    const int i = blockIdx.x * blockDim.x + threadIdx.x;
    if (i >= 640) return;
    const int b = i / 10, o = i % 10;
    const float* p  = pool + b * 64;
    const float* wr = w + o * 64;
    float acc = 0.0f;
#pragma unroll 8
    for (int c = 0; c < 64; ++c) acc += fabsf(p[c] - wr[c]);
    out[b * 10 + o] = -acc;
}

// ---------------------------------------------------------------------------
// Host orchestration
// ---------------------------------------------------------------------------
struct Blk { const float *w1, *g1, *b1, *w2, *g2, *b2, *wd, *gd, *bd; };

static inline int ilog2i(int v) { int s = 0; while ((1 << s) < v) ++s; return s; }

extern "C" void kernel_launch(void* const* d_in, const int* in_sizes, int n_in,
                              void* d_out, int out_size, void* d_ws, size_t ws_size,
                              hipStream_t stream) {
    (void)n_in; (void)out_size; (void)ws_size;
    // Detect flattening order: insertion order puts x (196608) first;
    // jax-sorted order puts bn1_b (16) first and x last (index 66).
    const bool ins = (in_sizes[0] == 64 * 3 * 32 * 32);
    auto F = [&](int i) -> const float* { return (const float*)d_in[i]; };
    auto P = [&](int i_ins, int i_sort) -> const float* { return F(ins ? i_ins : i_sort); };

    const float* x       = P(0, 66);
    const float* conv1_w = P(1, 4);
    const float* bn1_g   = P(2, 1);
    const float* bn1_b   = P(3, 0);
    const float* fc_w    = P(64, 5);
    const float* bn2_g   = P(65, 3);
    const float* bn2_b   = P(66, 2);

    auto mkB = [&](int bi, int bs, bool ds) -> Blk {
        Blk k = {};
        if (ins) {
            k.w1 = F(bi + 0); k.g1 = F(bi + 1); k.b1 = F(bi + 2);
            k.w2 = F(bi + 3); k.g2 = F(bi + 4); k.b2 = F(bi + 5);
            if (ds) { k.wd = F(bi + 6); k.gd = F(bi + 7); k.bd = F(bi + 8); }
        } else if (ds) {
            k.b1 = F(bs + 0); k.b2 = F(bs + 1); k.bd = F(bs + 2);
            k.g1 = F(bs + 3); k.g2 = F(bs + 4); k.gd = F(bs + 5);
            k.w1 = F(bs + 6); k.w2 = F(bs + 7); k.wd = F(bs + 8);
        } else {
            k.b1 = F(bs + 0); k.b2 = F(bs + 1); k.g1 = F(bs + 2);
            k.g2 = F(bs + 3); k.w1 = F(bs + 4); k.w2 = F(bs + 5);
        }
        return k;
    };
    Blk L1[3] = { mkB(4, 6, false),  mkB(10, 12, false), mkB(16, 18, false) };
    Blk L2[3] = { mkB(22, 24, true), mkB(31, 33, false), mkB(37, 39, false) };
    Blk L3[3] = { mkB(43, 45, true), mkB(52, 54, false), mkB(58, 60, false) };

    // Workspace layout (floats)
    float* wsf   = (float*)d_ws;
    float* stats = wsf;               // 128 used
    float* part  = wsf + 128;         // up to 4 groups * 8 slices * 32 = 1024
    float* pool  = wsf + 1280;        // 4096
    float* fco   = wsf + 1280 + 4096; // 640
    float* bufs[4];
    bufs[0] = wsf + 8192;
    bufs[1] = bufs[0] + (1 << 20);
    bufs[2] = bufs[1] + (1 << 20);
    bufs[3] = bufs[2] + (1 << 20);

    const int B = 64;
    auto conv = [&](const float* i, const float* w, float* o, int Cin, int H,
                    int Cout, int k, int s, int p, int Ho) {
        const int HWo = Ho * Ho;
        const int threads = (HWo < 256) ? HWo : 256;
        adder_conv<4><<<dim3(B * (Cout / 4)), dim3(threads), 0, stream>>>(
            i, w, o, Cin, H, H, Cout, k, k, s, p, Ho, Ho, ilog2i(Ho));
    };
    auto bnstats = [&](const float* i, int C, int HW) {
        if (HW >= 4) {
            bn_stats_wmma_vec<<<dim3(C / 16, 8), dim3(256), 0, stream>>>(i, part, C, ilog2i(HW));
            bn_stats_reduce<<<dim3(1), dim3(64), 0, stream>>>(part, stats, C, 8);
        } else {
            bn_stats_wmma_small<<<dim3((C + 15) / 16), dim3(256), 0, stream>>>(i, part, C, B * HW);
            bn_stats_reduce<<<dim3(1), dim3(64), 0, stream>>>(part, stats, C, 1);
        }
    };
    auto bnapp = [&](const float* i, float* o, const float* g, const float* bb,
                     int C, int HW, int mode, const float* res) {
        const int total = B * C * HW;
        const int cMask = ((C & (C - 1)) == 0) ? (C - 1) : -1;
        bn_apply<<<dim3((total + 255) / 256), dim3(256), 0, stream>>>(
            i, o, stats, g, bb, C, cMask, ilog2i(HW), mode, res,
            1.0f / (float)(B * HW), total);
    };

    // conv1 + BN + ReLU
    conv(x, conv1_w, bufs[0], 3, 32, 16, 3, 1, 1, 32);
    bnstats(bufs[0], 16, 1024);
    bnapp(bufs[0], bufs[0], bn1_g, bn1_b, 16, 1024, 0, nullptr);
    int hi = 0;

    auto step = [&](const Blk& k, int Cin, int Cout, int Hin, int s) {
        float* oth[3]; int m = 0;
        for (int i = 0; i < 4; ++i) if (i != hi) oth[m++] = bufs[i];
        float* t1 = oth[0]; float* t2 = oth[1]; float* t3 = oth[2];
        const float* h = bufs[hi];
        const int Ho = (Hin - 1) / s + 1;
        const int HW = Ho * Ho;
        // adder3x3 -> BN -> ReLU
        conv(h, k.w1, t1, Cin, Hin, Cout, 3, s, 1, Ho);
        bnstats(t1, Cout, HW);
        bnapp(t1, t1, k.g1, k.b1, Cout, HW, 0, nullptr);
        // identity path (downsample: adder1x1 -> BN)
        const float* ident = h;
        if (k.wd) {
            conv(h, k.wd, t3, Cin, Hin, Cout, 1, s, 0, Ho);
            bnstats(t3, Cout, HW);
            bnapp(t3, t3, k.gd, k.bd, Cout, HW, 1, nullptr);
            ident = t3;
        }
        // adder3x3 -> BN -> +identity -> ReLU
        conv(t1, k.w2, t2, Cout, Ho, Cout, 3, 1, 1, Ho);
        bnstats(t2, Cout, HW);
        bnapp(t2, t2, k.g2, k.b2, Cout, HW, 2, ident);
        for (int i = 0; i < 4; ++i) if (bufs[i] == t2) hi = i;
    };

    step(L1[0], 16, 16, 32, 1); step(L1[1], 16, 16, 32, 1); step(L1[2], 16, 16, 32, 1);
    step(L2[0], 16, 32, 32, 2); step(L2[1], 32, 32, 16, 1); step(L2[2], 32, 32, 16, 1);
    step(L3[0], 32, 64, 16, 2); step(L3[1], 64, 64,  8, 1); step(L3[2], 64, 64,  8, 1);

    // avgpool 8x8 -> [64,64]
    avgpool_k<<<dim3(16), dim3(256), 0, stream>>>(bufs[hi], pool, 4096);
    // adder FC 64 -> 10
    fc_adder<<<dim3(3), dim3(256), 0, stream>>>(pool, fc_w, fco);
    // final BN (C=10, N=64) -> d_out
    bnstats(fco, 10, 1);
    bnapp(fco, (float*)d_out, bn2_g, bn2_b, 10, 1, 1, nullptr);
}